// CropCrossEntropy_86371792322845
// MI455X (gfx1250) — compile-verified
//
#include <hip/hip_runtime.h>

typedef float v2f __attribute__((ext_vector_type(2)));
typedef float v4f __attribute__((ext_vector_type(4)));
typedef float v8f __attribute__((ext_vector_type(8)));

#define NIMG             512          // B*R = 32*16
#define PIX_PER_IMG      65536        // 256*256
#define SPLIT            4            // blocks per image (64 rows each)
#define THREADS          256          // 8 wave32s per block
#define CHUNKS_PER_BLOCK 4096         // float4 chunks per block (16384 px)
#define ITERS            (CHUNKS_PER_BLOCK / THREADS)   // 16
#define NBLOCKS          (NIMG * SPLIT)                 // 2048
#define NTOTAL_F         33554432.0f
#define LN2F             0.69314718055994530942f

// Pass 1: per-block partial sums of log2(t), t = inside-box ? p : 1-p.
__global__ __launch_bounds__(THREADS) void cce_partial(const float* __restrict__ mp,
                                                       const int* __restrict__ pg,
                                                       float* __restrict__ partial) {
    const int blk  = blockIdx.x;
    const int img  = blk >> 2;          // which (b,r) image
    const int part = blk & 3;           // which 64-row slab
    const int tid  = threadIdx.x;

    // box bounds: uniform per block -> scalar loads
    const int y0 = pg[4 * img + 0];
    const int x0 = pg[4 * img + 1];
    const int y1 = pg[4 * img + 2];
    const int x1 = pg[4 * img + 3];

    const float* base = mp + (size_t)img * PIX_PER_IMG + (size_t)part * (CHUNKS_PER_BLOCK * 4);

    float acc = 0.0f;  // sum of log2(t)
#pragma unroll
    for (int i = 0; i < ITERS; ++i) {
        const int c    = i * THREADS + tid;          // float4 chunk id within slab
        const v4f  v   = __builtin_nontemporal_load((const v4f*)(base + (size_t)c * 4));
        const int  row = (part << 6) + (c >> 6);
        const int  cb  = (c & 63) << 2;
        const bool rowin = (row >= y0) & (row <= y1);
#pragma unroll
        for (int j = 0; j < 4; ++j) {
            const int  col    = cb + j;
            const bool inside = rowin & (col >= x0) & (col <= x1);
            const float t     = inside ? v[j] : 1.0f - v[j];
            acc += __builtin_amdgcn_logf(t);         // raw v_log_f32 (log2)
        }
    }

    // ---- wave32 reduction via V_WMMA_F32_16X16X4_F32 ----
    // A[m,0]=p_m (lanes 0-15), A[m,2]=p_{m+16} (lanes 16-31), A[:,1]=A[:,3]=0.
    // B = all ones (layout-independent). D[m,n] = p_m + p_{m+16}.
    v2f a;     a.x = acc;  a.y = 0.0f;
    v2f bones; bones.x = 1.0f; bones.y = 1.0f;
    v8f cz = {};
    v8f d = __builtin_amdgcn_wmma_f32_16x16x4_f32(false, a, false, bones,
                                                  (short)0, cz, false, false);
    float s = d[0] + d[1] + d[2] + d[3] + d[4] + d[5] + d[6] + d[7];
    // lower half lanes hold sum over M=0..7, upper half over M=8..15; combine:
    s += __shfl_xor(s, 16, 32);

    __shared__ float wsum[THREADS / 32];
    const int lane = tid & 31;
    const int wave = tid >> 5;
    if (lane == 0) wsum[wave] = s;
    __syncthreads();
    if (tid == 0) {
        float t = 0.0f;
#pragma unroll
        for (int w = 0; w < THREADS / 32; ++w) t += wsum[w];
        partial[blk] = t;
    }
}

// Pass 2: deterministic fixed-order tree over 2048 partials; scale + negate.
__global__ __launch_bounds__(256) void cce_final(const float* __restrict__ partial,
                                                 float* __restrict__ out) {
    __shared__ float sm[256];
    const int tid = threadIdx.x;
    float t = 0.0f;
#pragma unroll
    for (int k = 0; k < NBLOCKS / 256; ++k) t += partial[tid + k * 256];
    sm[tid] = t;
    __syncthreads();
    for (int off = 128; off > 0; off >>= 1) {
        if (tid < off) sm[tid] += sm[tid + off];
        __syncthreads();
    }
    if (tid == 0) out[0] = -(LN2F * sm[0]) / NTOTAL_F;
}

extern "C" void kernel_launch(void* const* d_in, const int* in_sizes, int n_in,
                              void* d_out, int out_size, void* d_ws, size_t ws_size,
                              hipStream_t stream) {
    const float* mp = (const float*)d_in[0];   // mask_pred, 33,554,432 f32
    const int*   pg = (const int*)d_in[1];     // pos_gt, 512*4 ints
    float* partial  = (float*)d_ws;            // 2048 f32 scratch
    float* out      = (float*)d_out;

    cce_partial<<<NBLOCKS, THREADS, 0, stream>>>(mp, pg, partial);
    cce_final<<<1, 256, 0, stream>>>(partial, out);
}